// AdhocEGraph_60120952209875
// MI455X (gfx1250) — compile-verified
//
#include <hip/hip_runtime.h>
#include <hip/hip_bf16.h>

typedef __attribute__((ext_vector_type(2))) float v2f;
typedef __attribute__((ext_vector_type(8))) float v8f;

#define HDIM 64

// ---------------------------------------------------------------------------
// Workspace layout (floats), C read on device from num_classes scalar:
//   class_sum : B*C*64
//   node_sum  : B*N*64
//   cnt_c     : C
//   cnt_n     : N
// ---------------------------------------------------------------------------

__global__ void zero_ws_kernel(float* __restrict__ ws, const int* __restrict__ pC,
                               int B, int N, size_t cap) {
    const int C = *pC;
    size_t total = (size_t)B * C * HDIM + (size_t)B * N * HDIM + (size_t)C + (size_t)N;
    if (total > cap) total = cap;
    const size_t stride = (size_t)gridDim.x * blockDim.x;
    for (size_t i = (size_t)blockIdx.x * blockDim.x + threadIdx.x; i < total; i += stride)
        ws[i] = 0.0f;
}

// class2node mean, stage 1: sums + member counts (c2n_col == arange(N))
__global__ void class_agg_kernel(const float* __restrict__ emb,
                                 const int* __restrict__ c2n_row,
                                 float* __restrict__ ws, const int* __restrict__ pC,
                                 int B, int N) {
    const int C = *pC;
    const int idx = blockIdx.x * blockDim.x + threadIdx.x;
    if (idx >= N * HDIM) return;
    const int n = idx >> 6;
    const int h = idx & 63;
    const int c = c2n_row[n];
    float* class_sum = ws;
    float* cnt_c = ws + (size_t)B * C * HDIM + (size_t)B * N * HDIM;
    if (h == 0) atomicAdd(&cnt_c[c], 1.0f);
    for (int b = 0; b < B; ++b)
        atomicAdd(&class_sum[((size_t)b * C + c) * HDIM + h],
                  emb[((size_t)b * N + n) * HDIM + h]);
}

// class2node mean, stage 2: divide by max(count,1) in place
__global__ void class_norm_kernel(float* __restrict__ ws, const int* __restrict__ pC,
                                  int B, int N) {
    const int C = *pC;
    float* class_sum = ws;
    const float* cnt_c = ws + (size_t)B * C * HDIM + (size_t)B * N * HDIM;
    const size_t total = (size_t)B * C * HDIM;
    const size_t stride = (size_t)gridDim.x * blockDim.x;
    for (size_t i = (size_t)blockIdx.x * blockDim.x + threadIdx.x; i < total; i += stride) {
        const size_t c = (i >> 6) % (size_t)C;
        class_sum[i] *= 1.0f / fmaxf(cnt_c[c], 1.0f);
    }
}

// node2class mean, stage 1: gather class means along edges, sum per node + counts
__global__ void edge_agg_kernel(const int* __restrict__ n2c_row,
                                const int* __restrict__ n2c_col,
                                float* __restrict__ ws, const int* __restrict__ pC,
                                int B, int N, int E) {
    const int C = *pC;
    const int idx = blockIdx.x * blockDim.x + threadIdx.x;
    if (idx >= E * HDIM) return;
    const int e = idx >> 6;
    const int h = idx & 63;
    const int rn = n2c_row[e];
    const int cc = n2c_col[e];
    const float* class_emb = ws;
    float* node_sum = ws + (size_t)B * C * HDIM;
    float* cnt_n = node_sum + (size_t)B * N * HDIM + (size_t)C;
    if (h == 0) atomicAdd(&cnt_n[rn], 1.0f);
    for (int b = 0; b < B; ++b)
        atomicAdd(&node_sum[((size_t)b * N + rn) * HDIM + h],
                  class_emb[((size_t)b * C + cc) * HDIM + h]);
}

// ---------------------------------------------------------------------------
// Fused: x = emb + node_sum/max(cnt,1); h = x@W1+b1; LN; ReLU; out = h@W2+b2
// One wave per 16-row tile. 64 x V_WMMA_F32_16X16X4_F32 per tile.
// A 16x4 layout: lane(0-15)=M, vgpr0/1 = K{0,1}; lanes 16-31 = K{2,3}.
// B 4x16 layout (mirror): lane&15=N, vgpr0/1 = K{0,1}; lanes>=16 = K{2,3}.
// C/D 16x16:  N = lane&15, M = vgpr + 8*(lane>=16).
//
// W1 is staged in LDS pre-swizzled as row-pairs:
//   lds_w1[(q*64 + col)*2 + {0,1}] = W1[2q][col], W1[2q+1][col]   (q = 0..31)
// so each B-fragment is a single 8-byte ds_load_b64 (bank-conflict free).
// ---------------------------------------------------------------------------
__global__ void __launch_bounds__(256)
fused_mlp_wmma_kernel(const float* __restrict__ emb,
                      const float* __restrict__ W1g,   // [64][64] row k, col n
                      const float* __restrict__ b1g,
                      const float* __restrict__ gammag,
                      const float* __restrict__ betag,
                      const float* __restrict__ W2g,
                      const float* __restrict__ b2g,
                      const float* __restrict__ ws, const int* __restrict__ pC,
                      float* __restrict__ out,
                      int total_rows, int B, int N) {
    __shared__ float lds_w1[HDIM * HDIM];     // swizzled row-pair layout
    __shared__ float lds_b1[HDIM], lds_g[HDIM], lds_bt[HDIM], lds_w2[HDIM];

    // swizzled staging: pair rows (2q, 2q+1) interleaved per column
    for (int i = threadIdx.x; i < 32 * HDIM; i += blockDim.x) {
        const int q   = i >> 6;     // 0..31
        const int col = i & 63;
        lds_w1[2 * i]     = W1g[(2 * q) * HDIM + col];
        lds_w1[2 * i + 1] = W1g[(2 * q + 1) * HDIM + col];
    }
    if (threadIdx.x < HDIM) {
        lds_b1[threadIdx.x] = b1g[threadIdx.x];
        lds_g[threadIdx.x]  = gammag[threadIdx.x];
        lds_bt[threadIdx.x] = betag[threadIdx.x];
        lds_w2[threadIdx.x] = W2g[threadIdx.x];
    }
    __syncthreads();

    const int C = *pC;
    const float* node_sum = ws + (size_t)B * C * HDIM;
    const float* cnt_n    = node_sum + (size_t)B * N * HDIM + (size_t)C;

    const int lane = threadIdx.x & 31;
    const int wave = threadIdx.x >> 5;
    const int tile = blockIdx.x * (blockDim.x >> 5) + wave;
    const int row0 = tile * 16;
    if (row0 >= total_rows) return;           // wave-uniform: EXEC stays all-1

    const int lo = lane & 15;
    const int hi = lane >> 4;
    const int r  = row0 + lo;                 // row this lane feeds into A-frags
    const float invc = 1.0f / fmaxf(cnt_n[r % N], 1.0f);

    const float* xe = emb + (size_t)r * HDIM;
    const float* xs = node_sum + (size_t)r * HDIM;

    v8f acc0 = {}, acc1 = {}, acc2 = {}, acc3 = {};

#pragma unroll
    for (int kc = 0; kc < 16; ++kc) {
        const int k0 = kc * 4 + hi * 2;       // K pair handled by this half-wave
        const v2f ev = *(const v2f*)(xe + k0);
        const v2f sv = *(const v2f*)(xs + k0);
        const v2f a = ev + sv * invc;

        const int q = kc * 2 + hi;            // row-pair index = k0/2
        const float* wbase = lds_w1 + ((q * HDIM + lo) << 1);
        const v2f bf0 = *(const v2f*)(wbase);        // cols  0..15
        const v2f bf1 = *(const v2f*)(wbase + 32);   // cols 16..31
        const v2f bf2 = *(const v2f*)(wbase + 64);   // cols 32..47
        const v2f bf3 = *(const v2f*)(wbase + 96);   // cols 48..63

        acc0 = __builtin_amdgcn_wmma_f32_16x16x4_f32(false, a, false, bf0, (short)0, acc0, false, false);
        acc1 = __builtin_amdgcn_wmma_f32_16x16x4_f32(false, a, false, bf1, (short)0, acc1, false, false);
        acc2 = __builtin_amdgcn_wmma_f32_16x16x4_f32(false, a, false, bf2, (short)0, acc2, false, false);
        acc3 = __builtin_amdgcn_wmma_f32_16x16x4_f32(false, a, false, bf3, (short)0, acc3, false, false);
    }

    // Epilogue: per-lane column n = nt*16+lo; rows M = m + 8*hi live in acc[m].
    const float a0 = lds_b1[lo], a1 = lds_b1[16 + lo], a2 = lds_b1[32 + lo], a3 = lds_b1[48 + lo];
    const float g0 = lds_g[lo],  g1 = lds_g[16 + lo],  g2 = lds_g[32 + lo],  g3 = lds_g[48 + lo];
    const float t0 = lds_bt[lo], t1 = lds_bt[16 + lo], t2 = lds_bt[32 + lo], t3 = lds_bt[48 + lo];
    const float u0 = lds_w2[lo], u1 = lds_w2[16 + lo], u2 = lds_w2[32 + lo], u3 = lds_w2[48 + lo];
    const float bias2 = b2g[0];

    float logit[8];
#pragma unroll
    for (int m = 0; m < 8; ++m) {
        float d0 = acc0[m] + a0;
        float d1 = acc1[m] + a1;
        float d2 = acc2[m] + a2;
        float d3 = acc3[m] + a3;

        float s = d0 + d1 + d2 + d3;                 // reduce over 64 cols (16 lanes x 4)
        s += __shfl_xor(s, 1); s += __shfl_xor(s, 2);
        s += __shfl_xor(s, 4); s += __shfl_xor(s, 8);
        const float mu = s * 0.015625f;

        float q = (d0 - mu) * (d0 - mu) + (d1 - mu) * (d1 - mu)
                + (d2 - mu) * (d2 - mu) + (d3 - mu) * (d3 - mu);
        q += __shfl_xor(q, 1); q += __shfl_xor(q, 2);
        q += __shfl_xor(q, 4); q += __shfl_xor(q, 8);
        const float rstd = rsqrtf(q * 0.015625f + 1e-5f);

        const float h0 = fmaxf((d0 - mu) * rstd * g0 + t0, 0.0f);
        const float h1 = fmaxf((d1 - mu) * rstd * g1 + t1, 0.0f);
        const float h2 = fmaxf((d2 - mu) * rstd * g2 + t2, 0.0f);
        const float h3 = fmaxf((d3 - mu) * rstd * g3 + t3, 0.0f);

        float l = h0 * u0 + h1 * u1 + h2 * u2 + h3 * u3;
        l += __shfl_xor(l, 1); l += __shfl_xor(l, 2);
        l += __shfl_xor(l, 4); l += __shfl_xor(l, 8);
        logit[m] = l + bias2;
    }

    if (lo == 0) {
        float* o = out + row0 + hi * 8;
#pragma unroll
        for (int m = 0; m < 8; ++m) o[m] = logit[m];
    }
}

extern "C" void kernel_launch(void* const* d_in, const int* in_sizes, int n_in,
                              void* d_out, int out_size, void* d_ws, size_t ws_size,
                              hipStream_t stream) {
    const float* emb   = (const float*)d_in[0];
    const float* W1    = (const float*)d_in[1];
    const float* b1    = (const float*)d_in[2];
    const float* gamma = (const float*)d_in[3];
    const float* beta  = (const float*)d_in[4];
    const float* W2    = (const float*)d_in[5];
    const float* b2    = (const float*)d_in[6];
    const int* n2c_row = (const int*)d_in[7];
    const int* n2c_col = (const int*)d_in[8];
    const int* c2n_row = (const int*)d_in[9];
    // d_in[10] = c2n_col (== arange(N), implicit)
    const int* pC      = (const int*)d_in[11];   // num_classes, read on device

    const int E = in_sizes[7];
    const int N = in_sizes[9];
    const int B = in_sizes[0] / (N * HDIM);

    float* ws = (float*)d_ws;
    const size_t cap = ws_size / sizeof(float);

    zero_ws_kernel<<<2048, 256, 0, stream>>>(ws, pC, B, N, cap);

    class_agg_kernel<<<(N * HDIM + 255) / 256, 256, 0, stream>>>(emb, c2n_row, ws, pC, B, N);

    class_norm_kernel<<<4096, 256, 0, stream>>>(ws, pC, B, N);

    edge_agg_kernel<<<(E * HDIM + 255) / 256, 256, 0, stream>>>(n2c_row, n2c_col, ws, pC, B, N, E);

    const int total_rows = B * N;
    const int tiles = (total_rows + 15) / 16;
    const int waves_per_block = 256 / 32;
    const int blocks = (tiles + waves_per_block - 1) / waves_per_block;
    fused_mlp_wmma_kernel<<<blocks, 256, 0, stream>>>(emb, W1, b1, gamma, beta, W2, b2,
                                                      ws, pC, (float*)d_out,
                                                      total_rows, B, N);
}